// TemporalGNN_4784593567836
// MI455X (gfx1250) — compile-verified
//
#include <hip/hip_runtime.h>
#include <hip/hip_bf16.h>

// ---------------- problem constants (from reference) ----------------
#define TT 5
#define BB 512
#define NAg 16
#define SS 64
#define AA 16
#define HIDc 64
#define HH 4
#define FD 256            // HID*H
#define LAT 128
#define NN (BB*NAg)       // 8192
#define FIN (SS+AA)       // 80
#define NEG_SLOPE 0.2f

typedef __attribute__((ext_vector_type(2))) float v2f;
typedef __attribute__((ext_vector_type(8))) float v8f;

// ---------------- WMMA fp32 GEMM:  C[M,N] = A[M,K] @ W[N,K]^T (+bias)(+relu) ----
// Block = 128 threads (4 waves) owning a 64x64 C tile:
//   * 64xK weight tile staged in LDS once per block (pad to K+2 floats/row ->
//     stride-2 bank walk for the 16 column readers, 8B aligned for ds_load_b64)
//   * each wave computes 16x64 via 4 x V_WMMA_F32_16X16X4_F32 per k-step,
//     A fragment loaded once from global (1 global_load_b64) and reused;
//     B fragments come from LDS (DScnt, short latency) so WMMAs never have to
//     drain LOADcnt, letting the A-load stream run ahead.
// Fragment layouts per CDNA5 ISA:
//  A-frag 16x4: lanes 0-15 row M=l, VGPR0=K+0/VGPR1=K+1; lanes 16-31 K+2/K+3.
//  B-frag 4x16: lanes 0-15 col N=l, VGPR0=K+0/VGPR1=K+1; lanes 16-31 K+2/K+3.
//  D 16x16: lane<16 -> N=l, rows = vgpr 0..7; lane>=16 -> rows 8..15.
// Requires: M % 64 == 0, Nout % 64 == 0, K % 4 == 0 (true for all calls).
__global__ void k_gemm_wmma_f32(const float* __restrict__ A,
                                const float* __restrict__ W,
                                const float* __restrict__ bias,
                                float* __restrict__ C,
                                int M, int Nout, int K, int relu)
{
    extern __shared__ float sW[];              // [64][K+2]
    const int ldw = K + 2;
    const int tilesN = Nout >> 6;              // 64-wide macro-tiles
    const int tmg = blockIdx.x / tilesN;       // group of 4 row-tiles
    const int tn  = blockIdx.x % tilesN;

    // stage the 64 x K weight tile
    for (int i = threadIdx.x; i < 64 * K; i += blockDim.x) {
        int r = i / K, c = i % K;
        sW[r * ldw + c] = W[(size_t)(tn * 64 + r) * K + c];
    }
    __syncthreads();

    const int wave = threadIdx.x >> 5;
    const int lane = threadIdx.x & 31;
    const int tm  = tmg * 4 + wave;            // this wave's 16-row tile
    const int l15 = lane & 15;
    const int hi2 = (lane >> 4) << 1;          // 0 or 2

    const float* __restrict__ arow = A + (size_t)(tm * 16 + l15) * K;
    const float* __restrict__ w0 = sW + l15 * ldw;
    const float* __restrict__ w1 = w0 + 16 * ldw;
    const float* __restrict__ w2 = w0 + 32 * ldw;
    const float* __restrict__ w3 = w0 + 48 * ldw;

    v8f acc[4] = {v8f{}, v8f{}, v8f{}, v8f{}};
    for (int k = 0; k < K; k += 4) {
        const int kk = k + hi2;
        v2f af, b0, b1, b2, b3;
        af.x = arow[kk]; af.y = arow[kk + 1];  // 1 global_load_b64, reused 4x
        b0.x = w0[kk];   b0.y = w0[kk + 1];    // 4 ds_load_b64 from LDS
        b1.x = w1[kk];   b1.y = w1[kk + 1];
        b2.x = w2[kk];   b2.y = w2[kk + 1];
        b3.x = w3[kk];   b3.y = w3[kk + 1];
        acc[0] = __builtin_amdgcn_wmma_f32_16x16x4_f32(false, af, false, b0, (short)0, acc[0], false, false);
        acc[1] = __builtin_amdgcn_wmma_f32_16x16x4_f32(false, af, false, b1, (short)0, acc[1], false, false);
        acc[2] = __builtin_amdgcn_wmma_f32_16x16x4_f32(false, af, false, b2, (short)0, acc[2], false, false);
        acc[3] = __builtin_amdgcn_wmma_f32_16x16x4_f32(false, af, false, b3, (short)0, acc[3], false, false);
    }

    const int rbase = tm * 16 + ((lane >> 4) << 3);
    const int cbase = tn * 64 + l15;
#pragma unroll
    for (int j = 0; j < 4; ++j) {
        const int col = cbase + j * 16;
        const float bv = bias ? bias[col] : 0.0f;
#pragma unroll
        for (int i = 0; i < 8; ++i) {
            float v = acc[j][i] + bv;
            if (relu) v = v > 0.0f ? v : 0.0f;
            C[(size_t)(rbase + i) * Nout + col] = v;
        }
    }
}

// ---------------- node feature construction --------------------------------
// nodes[n, t, f] : f<S -> (agent==0 ? signals[t,b,f] : 0) ; else action feature
__global__ void k_build_nodes(const float* __restrict__ sig,
                              const float* __restrict__ acts,
                              float* __restrict__ nodes)
{
    int idx = blockIdx.x * blockDim.x + threadIdx.x;
    const int total = NN * TT * FIN;
    if (idx >= total) return;
    int f  = idx % FIN;
    int nt = idx / FIN;
    int t  = nt % TT;
    int n  = nt / TT;
    int b  = n / NAg, a = n % NAg;
    float v;
    if (f < SS) v = (a == 0) ? sig[(t * BB + b) * SS + f] : 0.0f;
    else        v = acts[((t * BB + b) * NAg + a) * AA + (f - SS)];
    nodes[idx] = v;
}

// ---------------- GAT attention logits (per node,t,head) --------------------
__global__ void k_att_scores(const float* __restrict__ xl,
                             const float* __restrict__ asrc,
                             const float* __restrict__ adst,
                             float* __restrict__ es, float* __restrict__ ed)
{
    int idx = blockIdx.x * blockDim.x + threadIdx.x;  // over N*T*H
    const int total = NN * TT * HH;
    if (idx >= total) return;
    int h  = idx % HH;
    int nt = idx / HH;
    const float* row = xl + (size_t)nt * FD + h * HIDc;
    const float* wsv = asrc + h * HIDc;
    const float* wdv = adst + h * HIDc;
    float s = 0.0f, d = 0.0f;
#pragma unroll 8
    for (int c = 0; c < HIDc; ++c) { float v = row[c]; s += v * wsv[c]; d += v * wdv[c]; }
    es[idx] = s; ed[idx] = d;
}

// ---------------- GAT dense per-batch aggregation ---------------------------
// Edge set = all-to-all + self-loops per batch of 16 agents, so segment
// softmax over dst is a dense softmax over the 16 sources.
__global__ void k_gat_aggregate(const float* __restrict__ xl,
                                const float* __restrict__ es,
                                const float* __restrict__ ed,
                                const float* __restrict__ bias,
                                float* __restrict__ out)
{
    const int bt = blockIdx.x;           // B*T blocks
    const int t = bt % TT, b = bt / TT;
    __shared__ float sAlpha[NAg][NAg][HH];     // [src][dst][h]
    __shared__ float sX[NAg][FD];
    const int tid = threadIdx.x;         // 256 threads

    // stage xl tile (16 nodes x 256 feats)
    for (int i = tid; i < NAg * FD; i += 256) {
        int sn = i / FD, f = i % FD;
        sX[sn][f] = xl[((size_t)((b * NAg + sn) * TT + t)) * FD + f];
    }
    // raw leaky-relu'd logits
    for (int i = tid; i < NAg * NAg * HH; i += 256) {
        int h = i % HH; int rem = i / HH;
        int dn = rem % NAg, sn = rem / NAg;
        float e = es[((b * NAg + sn) * TT + t) * HH + h]
                + ed[((b * NAg + dn) * TT + t) * HH + h];
        e = e > 0.0f ? e : NEG_SLOPE * e;
        sAlpha[sn][dn][h] = e;
    }
    __syncthreads();
    // softmax over the 16 sources for each (dst, head)
    if (tid < NAg * HH) {
        int h = tid % HH, dn = tid / HH;
        float m = -1e30f;
        for (int s2 = 0; s2 < NAg; ++s2) m = fmaxf(m, sAlpha[s2][dn][h]);
        float sum = 0.0f;
        for (int s2 = 0; s2 < NAg; ++s2) {
            float v = __expf(sAlpha[s2][dn][h] - m);
            sAlpha[s2][dn][h] = v; sum += v;
        }
        float inv = 1.0f / sum;
        for (int s2 = 0; s2 < NAg; ++s2) sAlpha[s2][dn][h] *= inv;
    }
    __syncthreads();
    // weighted sum + bias + relu ; thread owns one feature column
    const int f = tid;
    const int h = f / HIDc;
    const float bv = bias[f];
    for (int dn = 0; dn < NAg; ++dn) {
        float acc = 0.0f;
#pragma unroll
        for (int s2 = 0; s2 < NAg; ++s2) acc += sAlpha[s2][dn][h] * sX[s2][f];
        float v = acc + bv;
        v = v > 0.0f ? v : 0.0f;
        out[((size_t)((b * NAg + dn) * TT + t)) * FD + f] = v;
    }
}

// ---------------- gather seq (agent 0 of each batch) ------------------------
__global__ void k_gather_seq(const float* __restrict__ x, float* __restrict__ seq)
{
    int idx = blockIdx.x * blockDim.x + threadIdx.x;  // B*T*FD
    if (idx >= BB * TT * FD) return;
    int f = idx % FD; int bt = idx / FD;
    int t = bt % TT, b = bt / TT;
    seq[idx] = x[((size_t)((b * NAg) * TT + t)) * FD + f];
}

// ---------------- causal attention (T=5, per batch/head) --------------------
// NOTE: reference adds the causal mask as a +1.0 additive bias (not -inf).
__global__ void k_causal_attn(const float* __restrict__ qkv, float* __restrict__ ctx)
{
    const int bh = blockIdx.x;           // B*H blocks, 64 threads
    const int h = bh % HH, b = bh / HH;
    __shared__ float sQ[TT][HIDc], sK[TT][HIDc], sV[TT][HIDc], sS[TT][TT];
    const int tid = threadIdx.x;
    for (int i = tid; i < TT * HIDc; i += 64) {
        int tt = i / HIDc, d = i % HIDc;
        const float* base = qkv + (size_t)(b * TT + tt) * (3 * FD);
        sQ[tt][d] = base[h * HIDc + d];
        sK[tt][d] = base[FD + h * HIDc + d];
        sV[tt][d] = base[2 * FD + h * HIDc + d];
    }
    __syncthreads();
    if (tid < TT * TT) {
        int i = tid / TT, j = tid % TT;
        float s = 0.0f;
#pragma unroll 8
        for (int d = 0; d < HIDc; ++d) s += sQ[i][d] * sK[j][d];
        s = s * 0.125f + ((i >= j) ? 1.0f : 0.0f);   // 1/sqrt(64) + additive mask
        sS[i][j] = s;
    }
    __syncthreads();
    if (tid < TT) {
        int i = tid;
        float m = -1e30f;
        for (int j = 0; j < TT; ++j) m = fmaxf(m, sS[i][j]);
        float sum = 0.0f;
        for (int j = 0; j < TT; ++j) { float v = __expf(sS[i][j] - m); sS[i][j] = v; sum += v; }
        float inv = 1.0f / sum;
        for (int j = 0; j < TT; ++j) sS[i][j] *= inv;
    }
    __syncthreads();
    const int d = tid;  // 0..63
    for (int i = 0; i < TT; ++i) {
        float c = 0.0f;
#pragma unroll
        for (int j = 0; j < TT; ++j) c += sS[i][j] * sV[j][d];
        ctx[(size_t)(b * TT + i) * FD + h * HIDc + d] = c;
    }
}

// ---------------- gather last-timestep features -----------------------------
__global__ void k_gather_feat(const float* __restrict__ ctxo, float* __restrict__ feat)
{
    int idx = blockIdx.x * blockDim.x + threadIdx.x;  // B*FD
    if (idx >= BB * FD) return;
    int f = idx % FD, b = idx / FD;
    feat[idx] = ctxo[(size_t)(b * TT + (TT - 1)) * FD + f];
}

// ---------------- in-place row softmax over 64 cols -------------------------
__global__ void k_softmax64(float* __restrict__ p)
{
    int r = blockIdx.x * blockDim.x + threadIdx.x;
    if (r >= BB) return;
    float* row = p + (size_t)r * 64;
    float m = -1e30f;
    for (int j = 0; j < 64; ++j) m = fmaxf(m, row[j]);
    float s = 0.0f;
    for (int j = 0; j < 64; ++j) { float v = __expf(row[j] - m); row[j] = v; s += v; }
    float inv = 1.0f / s;
    for (int j = 0; j < 64; ++j) row[j] *= inv;
}

// ---------------------------------------------------------------------------
static inline int gemm_blocks(int M, int Nout) {
    return (M >> 6) * (Nout >> 6);          // 64x64 C tile per 4-wave block
}
static inline size_t gemm_lds(int K) {
    return (size_t)64 * (K + 2) * sizeof(float);
}

extern "C" void kernel_launch(void* const* d_in, const int* in_sizes, int n_in,
                              void* d_out, int out_size, void* d_ws, size_t ws_size,
                              hipStream_t stream) {
    (void)in_sizes; (void)n_in; (void)out_size; (void)ws_size;
    const float* signals   = (const float*)d_in[0];
    const float* nacts     = (const float*)d_in[1];
    const float* lin_w0    = (const float*)d_in[2];
    const float* att_src0  = (const float*)d_in[3];
    const float* att_dst0  = (const float*)d_in[4];
    const float* bias0     = (const float*)d_in[5];
    const float* lin_w1    = (const float*)d_in[6];
    const float* att_src1  = (const float*)d_in[7];
    const float* att_dst1  = (const float*)d_in[8];
    const float* bias1     = (const float*)d_in[9];
    const float* in_proj_w = (const float*)d_in[10];
    const float* in_proj_b = (const float*)d_in[11];
    const float* out_proj_w= (const float*)d_in[12];
    const float* out_proj_b= (const float*)d_in[13];
    const float* w_mean    = (const float*)d_in[14];
    const float* b_mean    = (const float*)d_in[15];
    const float* w_logvar  = (const float*)d_in[16];
    const float* b_logvar  = (const float*)d_in[17];
    const float* w_belief  = (const float*)d_in[18];
    const float* b_belief  = (const float*)d_in[19];
    // d_in[20] = edge_index (int64) — unused: graph is dense all-to-all+loops

    float* out = (float*)d_out;

    // workspace carve-up (floats)
    float* ws    = (float*)d_ws;
    float* nodes = ws;                            // N*T*FIN      = 3,276,800
    float* xl    = nodes + (size_t)NN*TT*FIN;     // N*T*FD       = 10,485,760
    float* xa    = xl    + (size_t)NN*TT*FD;      // N*T*FD
    float* es    = xa    + (size_t)NN*TT*FD;      // N*T*H
    float* ed    = es    + (size_t)NN*TT*HH;      // N*T*H
    float* seq   = ed    + (size_t)NN*TT*HH;      // B*T*FD
    float* qkv   = seq   + (size_t)BB*TT*FD;      // B*T*3FD
    float* ctx   = qkv   + (size_t)BB*TT*3*FD;    // B*T*FD
    float* ctxo  = ctx   + (size_t)BB*TT*FD;      // B*T*FD
    float* feat  = ctxo  + (size_t)BB*TT*FD;      // B*FD

    const int MT = NN * TT;        // 40960 rows for GAT GEMMs
    const int MS = BB * TT;        // 2560 rows for seq GEMMs

    // 0) node features
    k_build_nodes<<<(NN*TT*FIN + 255)/256, 256, 0, stream>>>(signals, nacts, nodes);

    // ---- GAT layer 0 ----
    k_gemm_wmma_f32<<<gemm_blocks(MT, FD), 128, gemm_lds(FIN), stream>>>(nodes, lin_w0, nullptr, xl, MT, FD, FIN, 0);
    k_att_scores<<<(MT*HH + 255)/256, 256, 0, stream>>>(xl, att_src0, att_dst0, es, ed);
    k_gat_aggregate<<<BB*TT, 256, 0, stream>>>(xl, es, ed, bias0, xa);

    // ---- GAT layer 1 ----
    k_gemm_wmma_f32<<<gemm_blocks(MT, FD), 128, gemm_lds(FD), stream>>>(xa, lin_w1, nullptr, xl, MT, FD, FD, 0);
    k_att_scores<<<(MT*HH + 255)/256, 256, 0, stream>>>(xl, att_src1, att_dst1, es, ed);
    k_gat_aggregate<<<BB*TT, 256, 0, stream>>>(xl, es, ed, bias1, xa);

    // ---- temporal attention over agent-0 sequence ----
    k_gather_seq<<<(BB*TT*FD + 255)/256, 256, 0, stream>>>(xa, seq);
    k_gemm_wmma_f32<<<gemm_blocks(MS, 3*FD), 128, gemm_lds(FD), stream>>>(seq, in_proj_w, in_proj_b, qkv, MS, 3*FD, FD, 0);
    k_causal_attn<<<BB*HH, 64, 0, stream>>>(qkv, ctx);
    k_gemm_wmma_f32<<<gemm_blocks(MS, FD), 128, gemm_lds(FD), stream>>>(ctx, out_proj_w, out_proj_b, ctxo, MS, FD, FD, 0);

    // ---- heads ----
    k_gather_feat<<<(BB*FD + 255)/256, 256, 0, stream>>>(ctxo, feat);
    float* out_mean   = out;
    float* out_logvar = out + (size_t)BB * LAT;
    float* out_belief = out + (size_t)BB * LAT * 2;
    k_gemm_wmma_f32<<<gemm_blocks(BB, LAT), 128, gemm_lds(FD), stream>>>(feat, w_mean,   b_mean,   out_mean,   BB, LAT, FD, 0);
    k_gemm_wmma_f32<<<gemm_blocks(BB, LAT), 128, gemm_lds(FD), stream>>>(feat, w_logvar, b_logvar, out_logvar, BB, LAT, FD, 0);
    k_gemm_wmma_f32<<<gemm_blocks(BB, SS),  128, gemm_lds(FD), stream>>>(feat, w_belief, b_belief, out_belief, BB, SS,  FD, 0);
    k_softmax64<<<(BB + 255)/256, 256, 0, stream>>>(out_belief);
}